// DifferentialRenderLoss_45440753992227
// MI455X (gfx1250) — compile-verified
//
#include <hip/hip_runtime.h>

#define N_CAM 2
#define PH 96
#define PW 144
#define NPTS 200
#define VD 32
#define VH 128
#define VW 384
#define VOXN (VD * VH * VW)           // 1,572,864
#define NRAYS (N_CAM * PH * PW)       // 27,648
#define EPSF 1e-10f
#define MIN_DEPTH_F 1.0f
#define MAX_DEPTH_F 4000.0f
#define DSTEP ((MAX_DEPTH_F - MIN_DEPTH_F) / (float)(NPTS - 1))
#define INV_VOXEL 0.4f                // 1 / 2.5

typedef __attribute__((ext_vector_type(2))) float v2f;
typedef __attribute__((ext_vector_type(8))) float v8f;

#if defined(__has_builtin)
#if __has_builtin(__builtin_amdgcn_wmma_f32_16x16x4_f32)
#define HAVE_WMMA_F32X4 1
#else
#define HAVE_WMMA_F32X4 0
#endif
#if __has_builtin(__builtin_amdgcn_global_load_async_to_lds_b32) && \
    __has_builtin(__builtin_amdgcn_s_wait_asynccnt)
#define HAVE_ASYNC_LDS 1
#else
#define HAVE_ASYNC_LDS 0
#endif
#else
#define HAVE_WMMA_F32X4 0
#define HAVE_ASYNC_LDS 0
#endif

// async-to-LDS builtin expects AS(1) int* (printed '__device__ int*') and
// AS(3) int* for the LDS side; b32 is an untyped bit move.
#define AS1P(p) ((__attribute__((address_space(1))) int*)((void*)(p)))
#define AS3P(p) ((__attribute__((address_space(3))) int*)((void*)(p)))

__device__ __forceinline__ float huber01(float x, float y) {
    float d = x - y;
    float t = fmaf(d * d, 100.0f, 1.0f);   // 1 + diff^2 / 0.1^2
    t = fmaxf(t, 1e-4f);
    return (sqrtf(t) - 1.0f) * 0.1f;
}

// ---------------------------------------------------------------------------
// Wave-wide sum via the matrix unit: D = A(16x4, lane values) x B(4x16, ones)
// Row sums land in D; lanes 0-15 hold rows 0-7, lanes 16-31 rows 8-15.
// Requires EXEC all-ones (call only from fully-converged code).
// ---------------------------------------------------------------------------
__device__ __forceinline__ float wave_sum32(float x) {
#if HAVE_WMMA_F32X4
    v2f a; a[0] = x;    a[1] = 0.0f;
    v2f b; b[0] = 1.0f; b[1] = 1.0f;          // ones: K-striping irrelevant
    v8f c = {0.f, 0.f, 0.f, 0.f, 0.f, 0.f, 0.f, 0.f};
    v8f d = __builtin_amdgcn_wmma_f32_16x16x4_f32(
        false, a, false, b, (short)0, c, false, false);
    float s = ((d[0] + d[1]) + (d[2] + d[3])) + ((d[4] + d[5]) + (d[6] + d[7]));
    s += __shfl_xor(s, 16, 32);               // combine row halves
    return s;
#else
    for (int m = 16; m > 0; m >>= 1) x += __shfl_xor(x, m, 32);
    return x;
#endif
}

// ---------------------------------------------------------------------------
// Fuse (density, r, g, b) planes into interleaved float4 volume (b128 loads).
// ---------------------------------------------------------------------------
__global__ __launch_bounds__(256) void fuse_kernel(
    const float* __restrict__ den, const float* __restrict__ col,
    float4* __restrict__ vol) {
    int i = blockIdx.x * 256 + threadIdx.x;
    vol[i] = make_float4(den[i], col[i], col[i + VOXN], col[i + 2 * VOXN]);
}

// ---------------------------------------------------------------------------
// One thread per ray: march 200 samples, trilinear sample, composite, loss.
// ---------------------------------------------------------------------------
template <bool FUSED>
__global__ __launch_bounds__(256) void render_kernel(
    const float4* __restrict__ vol,
    const float* __restrict__ den, const float* __restrict__ col,
    const float* __restrict__ tsil, const float* __restrict__ timg,
    const float* __restrict__ focal, const float* __restrict__ principal,
    const float* __restrict__ Rm, const float* __restrict__ Tm,
    float* __restrict__ out) {
    const int tid = threadIdx.x;
    const int ray = blockIdx.x * blockDim.x + tid;
    const int cam = ray / (PH * PW);
    const int pix = ray - cam * (PH * PW);
    const int py  = pix / PW;
    const int px  = pix - py * PW;

#if HAVE_ASYNC_LDS
    __shared__ float ls_sil[256];
    __shared__ float ls_img[256 * 3];
    // Stage the loss targets into LDS asynchronously; consumed after the march.
    __builtin_amdgcn_global_load_async_to_lds_b32(
        AS1P(tsil + ray), AS3P(&ls_sil[tid]), 0, 0);
    __builtin_amdgcn_global_load_async_to_lds_b32(
        AS1P(timg + ray * 3 + 0), AS3P(&ls_img[tid * 3 + 0]), 0, 0);
    __builtin_amdgcn_global_load_async_to_lds_b32(
        AS1P(timg + ray * 3 + 1), AS3P(&ls_img[tid * 3 + 1]), 0, 0);
    __builtin_amdgcn_global_load_async_to_lds_b32(
        AS1P(timg + ray * 3 + 2), AS3P(&ls_img[tid * 3 + 2]), 0, 0);
#endif

    const float* Rc = Rm + cam * 9;
    const float dcx = ((float)px + 0.5f - principal[cam * 2 + 0]) / focal[cam * 2 + 0];
    const float dcy = ((float)py + 0.5f - principal[cam * 2 + 1]) / focal[cam * 2 + 1];

    // dirs_world[j] = sum_k dirs_cam[k] * R[j][k]
    const float dwx = Rc[0] * dcx + Rc[1] * dcy + Rc[2];
    const float dwy = Rc[3] * dcx + Rc[4] * dcy + Rc[5];
    const float dwz = Rc[6] * dcx + Rc[7] * dcy + Rc[8];

    const float Tx = Tm[cam * 3 + 0], Ty = Tm[cam * 3 + 1], Tz = Tm[cam * 3 + 2];
    const float ox = -(Rc[0] * Tx + Rc[1] * Ty + Rc[2] * Tz);
    const float oy = -(Rc[3] * Tx + Rc[4] * Ty + Rc[5] * Tz);
    const float oz = -(Rc[6] * Tx + Rc[7] * Ty + Rc[8] * Tz);

    // grid coords: g = pos / VOXEL + (dim-1)/2   (VOL_TRANS == 0)
    const float gx0 = fmaf(fmaf(MIN_DEPTH_F, dwx, ox), INV_VOXEL, 0.5f * (VW - 1));
    const float gy0 = fmaf(fmaf(MIN_DEPTH_F, dwy, oy), INV_VOXEL, 0.5f * (VH - 1));
    const float gz0 = fmaf(fmaf(MIN_DEPTH_F, dwz, oz), INV_VOXEL, 0.5f * (VD - 1));
    const float dgx = DSTEP * dwx * INV_VOXEL;
    const float dgy = DSTEP * dwy * INV_VOXEL;
    const float dgz = DSTEP * dwz * INV_VOXEL;

    float transEps = 1.0f;   // cumprod(1 + EPS - d), shifted
    float prod1md  = 1.0f;   // prod(1 - d) for opacity
    float fr = 0.0f, fg = 0.0f, fb = 0.0f;

    for (int i = 0; i < NPTS; ++i) {
        const float fi = (float)i;
        const float gx = fmaf(fi, dgx, gx0);
        const float gy = fmaf(fi, dgy, gy0);
        const float gz = fmaf(fi, dgz, gz0);

        // prefetch the corner base 8 samples ahead (gfx1250 global_prefetch_b8)
        {
            const float pgx = fmaf(fi + 8.0f, dgx, gx0);
            const float pgy = fmaf(fi + 8.0f, dgy, gy0);
            const float pgz = fmaf(fi + 8.0f, dgz, gz0);
            if (pgx > 0.0f && pgx < (float)(VW - 1) &&
                pgy > 0.0f && pgy < (float)(VH - 1) &&
                pgz > 0.0f && pgz < (float)(VD - 1)) {
                const int pidx = (((int)pgz * VH + (int)pgy) * VW + (int)pgx);
                if (FUSED) __builtin_prefetch(vol + pidx, 0, 3);
                else       __builtin_prefetch(den + pidx, 0, 3);
            }
        }

        // sample entirely outside volume -> d == 0; transmittance still *= (1+EPS)
        if (gx <= -1.0f || gx >= (float)VW ||
            gy <= -1.0f || gy >= (float)VH ||
            gz <= -1.0f || gz >= (float)VD) {
            transEps *= (1.0f + EPSF);
            continue;
        }

        const float x0 = floorf(gx), y0 = floorf(gy), z0 = floorf(gz);
        const float fx = gx - x0, fy = gy - y0, fz = gz - z0;
        const int xi = (int)x0, yi = (int)y0, zi = (int)z0;

        float f0 = 0.0f, f1 = 0.0f, f2 = 0.0f, f3 = 0.0f;
#pragma unroll
        for (int dz = 0; dz < 2; ++dz)
#pragma unroll
            for (int dy = 0; dy < 2; ++dy)
#pragma unroll
                for (int dx = 0; dx < 2; ++dx) {
                    const int X = xi + dx, Y = yi + dy, Z = zi + dz;
                    if (X >= 0 && X < VW && Y >= 0 && Y < VH && Z >= 0 && Z < VD) {
                        const float w = (dx ? fx : 1.0f - fx) *
                                        (dy ? fy : 1.0f - fy) *
                                        (dz ? fz : 1.0f - fz);
                        const int idx = (Z * VH + Y) * VW + X;
                        if (FUSED) {
                            const float4 c = vol[idx];   // single global_load_b128
                            f0 = fmaf(w, c.x, f0);
                            f1 = fmaf(w, c.y, f1);
                            f2 = fmaf(w, c.z, f2);
                            f3 = fmaf(w, c.w, f3);
                        } else {
                            f0 = fmaf(w, den[idx], f0);
                            f1 = fmaf(w, col[idx], f1);
                            f2 = fmaf(w, col[idx + VOXN], f2);
                            f3 = fmaf(w, col[idx + 2 * VOXN], f3);
                        }
                    }
                }

        const float d = f0;
        const float w = d * transEps;          // weight = d * shifted cumprod
        fr = fmaf(w, f1, fr);
        fg = fmaf(w, f2, fg);
        fb = fmaf(w, f3, fb);
        prod1md  *= (1.0f - d);
        transEps *= (1.0f + EPSF - d);
    }

    const float opacity = 1.0f - prod1md;

#if HAVE_ASYNC_LDS
    __builtin_amdgcn_s_wait_asynccnt(0);       // own lane's staged data only
    const float tsv = ls_sil[tid];
    const float t0 = ls_img[tid * 3 + 0];
    const float t1 = ls_img[tid * 3 + 1];
    const float t2 = ls_img[tid * 3 + 2];
#else
    const float tsv = tsil[ray];
    const float t0 = timg[ray * 3 + 0];
    const float t1 = timg[ray * 3 + 1];
    const float t2 = timg[ray * 3 + 2];
#endif

    const float sil_l = fabsf(huber01(opacity, tsv));
    const float col_l = fabsf(huber01(fr, t0)) +
                        fabsf(huber01(fg, t1)) +
                        fabsf(huber01(fb, t2));

    // wave reduction through the matrix unit, then 8-wave LDS combine
    const float wc = wave_sum32(col_l);
    const float ws = wave_sum32(sil_l);

    __shared__ float wsum_col[8];
    __shared__ float wsum_sil[8];
    const int wid = tid >> 5, lane = tid & 31;
    if (lane == 0) { wsum_col[wid] = wc; wsum_sil[wid] = ws; }
    __syncthreads();
    if (tid == 0) {
        float c = 0.0f, s = 0.0f;
#pragma unroll
        for (int i = 0; i < 8; ++i) { c += wsum_col[i]; s += wsum_sil[i]; }
        atomicAdd(out + 0, c * (1.0f / (float)(NRAYS * 3)));
        atomicAdd(out + 1, s * (1.0f / (float)NRAYS));
    }
}

// ---------------------------------------------------------------------------
// bev_err = mean over (H,W) of |max over D of density|
// ---------------------------------------------------------------------------
__global__ __launch_bounds__(256) void bev_kernel(
    const float* __restrict__ den, float* __restrict__ out) {
    const int tid = threadIdx.x;
    const int idx = blockIdx.x * 256 + tid;   // y*VW + x, < VH*VW
    float m = den[idx];
#pragma unroll 4
    for (int z = 1; z < VD; ++z)
        m = fmaxf(m, den[z * (VH * VW) + idx]);
    const float wsum = wave_sum32(fabsf(m));

    __shared__ float wbuf[8];
    const int wid = tid >> 5, lane = tid & 31;
    if (lane == 0) wbuf[wid] = wsum;
    __syncthreads();
    if (tid == 0) {
        float s = 0.0f;
#pragma unroll
        for (int i = 0; i < 8; ++i) s += wbuf[i];
        atomicAdd(out + 2, s * (1.0f / (float)(VH * VW)));
    }
}

extern "C" void kernel_launch(void* const* d_in, const int* in_sizes, int n_in,
                              void* d_out, int out_size, void* d_ws, size_t ws_size,
                              hipStream_t stream) {
    const float* den   = (const float*)d_in[0];   // (1,1,VD,VH,VW)
    const float* col   = (const float*)d_in[1];   // (1,3,VD,VH,VW)
    const float* tsil  = (const float*)d_in[2];   // (N_CAM,PH,PW)
    const float* timg  = (const float*)d_in[3];   // (N_CAM,PH,PW,3)
    const float* focal = (const float*)d_in[4];   // (N_CAM,2)
    const float* prin  = (const float*)d_in[5];   // (N_CAM,2)
    const float* Rm    = (const float*)d_in[6];   // (N_CAM,3,3)
    const float* Tm    = (const float*)d_in[7];   // (N_CAM,3)
    float* out = (float*)d_out;                   // (color_err, sil_err, bev_err)

    (void)hipMemsetAsync(out, 0, 3 * sizeof(float), stream);

    const size_t fuse_bytes = (size_t)VOXN * sizeof(float4);  // ~25 MB
    if (ws_size >= fuse_bytes) {
        float4* vol = (float4*)d_ws;
        fuse_kernel<<<VOXN / 256, 256, 0, stream>>>(den, col, vol);
        render_kernel<true><<<NRAYS / 256, 256, 0, stream>>>(
            vol, den, col, tsil, timg, focal, prin, Rm, Tm, out);
    } else {
        render_kernel<false><<<NRAYS / 256, 256, 0, stream>>>(
            nullptr, den, col, tsil, timg, focal, prin, Rm, Tm, out);
    }
    bev_kernel<<<(VH * VW) / 256, 256, 0, stream>>>(den, out);
}